// MovingZScoreNorm_33311766348269
// MI455X (gfx1250) — compile-verified
//
#include <hip/hip_runtime.h>
#include <cstdint>
#include <cstddef>

// Moving z-score normalization, B=32 S=8192 C=128 fp32, W=24.
// Bandwidth-bound: stage tiles into LDS with the CDNA5 Tensor Data Mover
// (tensor_load_to_lds + s_wait_tensorcnt), then do all window reuse from LDS.

#define TILE_S 128                 // output rows per workgroup
#define HALO   24                  // >= W, rows of halo each side
#define ROWS   (TILE_S + 2*HALO)   // 176 LDS rows
#define CCH    128                 // channels
#define WIN    24
#define PADL   11                  // (W-1)/2
#define PADR   12                  // (W-1)-PADL
#define EPSV   1e-7f

typedef unsigned int v4u __attribute__((ext_vector_type(4)));
typedef int          v8i __attribute__((ext_vector_type(8)));
typedef int          v4i __attribute__((ext_vector_type(4)));

__global__ __launch_bounds__(256)
void mzsnorm_kernel(const float* __restrict__ x, float* __restrict__ out, int S) {
    __shared__ float lds_x[ROWS * CCH];   // 176*128*4 = 90112 B
    __shared__ float lds_sq[ROWS];        // per-row channel-summed sq dev

    const int b    = blockIdx.y;
    const int s0   = blockIdx.x * TILE_S;
    const int tid  = threadIdx.x;
    const int lane = tid & 31;
    const int wv   = tid >> 5;

    // valid global row range covered by this tile (+halo)
    const int g_lo    = max(0, s0 - HALO);
    const int g_hi    = min(S, s0 + TILE_S + HALO);
    const int n_rows  = g_hi - g_lo;
    const int l_first = g_lo - (s0 - HALO);   // first valid LDS row

    // ---- zero the invalid halo rows (boundary chunks only) ----
    {
        const int qpr  = CCH / 4;
        float4* q      = (float4*)lds_x;
        const int head = l_first * qpr;
        const int tail = (l_first + n_rows) * qpr;
        const int tot  = ROWS * qpr;
        float4 z; z.x = z.y = z.z = z.w = 0.0f;
        for (int i = tid; i < head; i += 256) q[i] = z;
        for (int i = tail + tid; i < tot; i += 256) q[i] = z;
    }
    __syncthreads();

    const float* gsrc = x + ((size_t)b * (size_t)S + (size_t)g_lo) * CCH;

#if __has_builtin(__builtin_amdgcn_tensor_load_to_lds) && __has_builtin(__builtin_amdgcn_s_wait_tensorcnt)
    // ---- TDM: DMA n_rows x 128 fp32 (row-contiguous) into LDS ----
    if (wv == 0) {
        const uint32_t lds_byte = (uint32_t)(uintptr_t)(&lds_x[l_first * CCH]);
        const uint64_t ga       = (uint64_t)(uintptr_t)gsrc;

        v4u g0;
        g0[0] = 1u;                                   // count=1, user descriptor
        g0[1] = lds_byte;                             // lds_addr (bytes)
        g0[2] = (uint32_t)ga;                         // global_addr[31:0]
        g0[3] = ((uint32_t)(ga >> 32) & 0x01FFFFFFu)  // global_addr[56:32]
              | 0x80000000u;                          // type=2 ("image")

        v8i g1;
        g1[0] = 0x00020000;                           // wg_mask=0, data_size=4B
        g1[1] = (int)(128u << 16);                    // tensor_dim0 = 128 (lo16 in hi half)
        g1[2] = (int)((uint32_t)n_rows << 16);        // tensor_dim1 = n_rows
        g1[3] = (int)(128u << 16);                    // tile_dim0 = 128
        g1[4] = (n_rows & 0xFFFF);                    // tile_dim1 = n_rows, tile_dim2=0
        g1[5] = 128;                                  // tensor_dim0_stride = 128 elems
        g1[6] = 0;
        g1[7] = 0;

        v4i g2; g2[0] = 0; g2[1] = 0; g2[2] = 0; g2[3] = 0;
        v4i g3; g3[0] = 0; g3[1] = 0; g3[2] = 0; g3[3] = 0;
        v8i g4; g4[0] = 0; g4[1] = 0; g4[2] = 0; g4[3] = 0;
               g4[4] = 0; g4[5] = 0; g4[6] = 0; g4[7] = 0;

        __builtin_amdgcn_tensor_load_to_lds(g0, g1, g2, g3, g4, 0);
        __builtin_amdgcn_s_wait_tensorcnt(0);
    }
#else
    // Fallback: cooperative vectorized copy.
    {
        const float4* gs = (const float4*)gsrc;
        float4* dst = (float4*)&lds_x[l_first * CCH];
        const int nq = n_rows * (CCH / 4);
        for (int i = tid; i < nq; i += 256) dst[i] = gs[i];
    }
#endif
    __syncthreads();

    // ---- Phase A: sq_tot[j] = sum_c (x[j,c]-mean[j,c])^2 for extended rows ----
    {
        const int EXTR = TILE_S + 2 * PADR;   // 152 rows: [s0-12, s0+T+12)
        const int perw = EXTR / 8;            // 19 rows per wave
        const int jbeg = s0 - PADR + wv * perw;
        const int cb   = lane * 4;

        float w0 = 0.f, w1 = 0.f, w2 = 0.f, w3 = 0.f;
        const int lr0 = jbeg - s0 + HALO;
        for (int r = lr0 - PADL; r <= lr0 + PADR; ++r) {
            const float* p = &lds_x[r * CCH + cb];
            w0 += p[0]; w1 += p[1]; w2 += p[2]; w3 += p[3];
        }
        for (int j = jbeg; j < jbeg + perw; ++j) {
            const int lr = j - s0 + HALO;
            if (j > jbeg) {   // slide: +row(j+12), -row(j-12)
                const float* pa = &lds_x[(lr + PADR) * CCH + cb];
                const float* ps = &lds_x[(lr - PADL - 1) * CCH + cb];
                w0 += pa[0] - ps[0]; w1 += pa[1] - ps[1];
                w2 += pa[2] - ps[2]; w3 += pa[3] - ps[3];
            }
            float sq = 0.f;
            if (j >= 0 && j < S) {
                const int cnt  = min(S, j + PADR + 1) - max(0, j - PADL);
                const float iv = 1.0f / (float)cnt;
                const float* p = &lds_x[lr * CCH + cb];
                const float d0 = p[0] - w0 * iv, d1 = p[1] - w1 * iv;
                const float d2 = p[2] - w2 * iv, d3 = p[3] - w3 * iv;
                sq = d0 * d0 + d1 * d1 + d2 * d2 + d3 * d3;
            }
            // wave32 butterfly reduce over the 32 lanes (128 channels total)
            for (int off = 16; off > 0; off >>= 1)
                sq += __shfl_xor(sq, off, 32);
            if (lane == 0) lds_sq[lr] = sq;
        }
    }
    __syncthreads();

    // ---- Phase B: var + normalize + store ----
    {
        const int perw = TILE_S / 8;      // 16 rows per wave
        const int sbeg = s0 + wv * perw;
        const int cb   = lane * 4;

        float w0 = 0.f, w1 = 0.f, w2 = 0.f, w3 = 0.f, vs = 0.f;
        const int lr0 = sbeg - s0 + HALO;
        for (int r = lr0 - PADL; r <= lr0 + PADR; ++r) {
            const float* p = &lds_x[r * CCH + cb];
            w0 += p[0]; w1 += p[1]; w2 += p[2]; w3 += p[3];
            vs += lds_sq[r];
        }
        float* ob = out + (size_t)b * (size_t)S * CCH;
        const float ivw = 1.0f / (float)(WIN - 1);
        for (int s = sbeg; s < sbeg + perw; ++s) {
            const int lr = s - s0 + HALO;
            if (s > sbeg) {
                const float* pa = &lds_x[(lr + PADR) * CCH + cb];
                const float* ps = &lds_x[(lr - PADL - 1) * CCH + cb];
                w0 += pa[0] - ps[0]; w1 += pa[1] - ps[1];
                w2 += pa[2] - ps[2]; w3 += pa[3] - ps[3];
                vs += lds_sq[lr + PADR] - lds_sq[lr - PADL - 1];
            }
            const int cnt   = min(S, s + PADR + 1) - max(0, s - PADL);
            const float iv  = 1.0f / (float)cnt;
            const float var = vs * ivw;
            const float is  = 1.0f / (sqrtf(var) + EPSV);
            const float* p  = &lds_x[lr * CCH + cb];
            float4 o;
            o.x = (p[0] - w0 * iv) * is;
            o.y = (p[1] - w1 * iv) * is;
            o.z = (p[2] - w2 * iv) * is;
            o.w = (p[3] - w3 * iv) * is;
            *(float4*)&ob[(size_t)s * CCH + cb] = o;
        }
    }
}

extern "C" void kernel_launch(void* const* d_in, const int* in_sizes, int n_in,
                              void* d_out, int out_size, void* d_ws, size_t ws_size,
                              hipStream_t stream) {
    const float* x = (const float*)d_in[0];
    float* out     = (float*)d_out;
    (void)d_ws; (void)ws_size; (void)n_in; (void)in_sizes;

    const int B = 32, C = 128;
    const int S = out_size / (B * C);         // 8192
    dim3 grid(S / TILE_S, B);                 // (64, 32)
    mzsnorm_kernel<<<grid, 256, 0, stream>>>(x, out, S);
}